// NTXentLoss_51350628991193
// MI455X (gfx1250) — compile-verified
//
#include <hip/hip_runtime.h>
#include <math.h>

typedef __attribute__((ext_vector_type(2))) float v2f;
typedef __attribute__((ext_vector_type(8))) float v8f;
typedef __attribute__((ext_vector_type(4))) int   v4i;

#define BHALF 4096          // B in the reference
#define NTOT  8192          // 2*B
#define DDIM  128
#define NTILE (NTOT / 16)   // 512 column/row tiles
#define WAVES 4             // waves per block, one 16-row tile each
#define TPB   (WAVES * 32)  // 128 threads
#define LSTRIDE 132         // padded LDS row stride (floats) -> conflict-free b64 reads

// ---- CDNA5 async global->LDS copy (no data VGPRs, tracked by ASYNCcnt) ----
__device__ __forceinline__ void async_copy_b128(const float* gsrc, float* ldst) {
#if __has_builtin(__builtin_amdgcn_global_load_async_to_lds_b128)
  __builtin_amdgcn_global_load_async_to_lds_b128(
      (__attribute__((address_space(1))) v4i*)(uintptr_t)gsrc,
      (__attribute__((address_space(3))) v4i*)(uintptr_t)ldst,
      /*imm offset=*/0, /*cpol=*/0);
#else
  unsigned l = (unsigned)(uintptr_t)ldst;   // LDS byte offset (addr[31:0])
  asm volatile("global_load_async_to_lds_b128 %0, %1, off"
               :: "v"(l), "v"(gsrc) : "memory");
#endif
}

__device__ __forceinline__ void wait_async0() {
#if __has_builtin(__builtin_amdgcn_s_wait_asynccnt)
  __builtin_amdgcn_s_wait_asynccnt(0);
#else
  asm volatile("s_wait_asynccnt 0x0" ::: "memory");
#endif
}

// ---------------------------------------------------------------------------
// Kernel 1: row L2-normalize  zn[r] = z[r] / max(||z[r]||, 1e-8)
// ---------------------------------------------------------------------------
__global__ __launch_bounds__(256) void ntxent_normalize(
    const float* __restrict__ zi, const float* __restrict__ zj,
    float* __restrict__ zn) {
  const int row  = (blockIdx.x * blockDim.x + threadIdx.x) >> 5;
  const int lane = threadIdx.x & 31;
  if (row >= NTOT) return;  // wave-uniform
  const float* src = (row < BHALF) ? (zi + (size_t)row * DDIM)
                                   : (zj + (size_t)(row - BHALF) * DDIM);
  float4 v = ((const float4*)src)[lane];
  float ss = v.x * v.x + v.y * v.y + v.z * v.z + v.w * v.w;
  #pragma unroll
  for (int m = 16; m >= 1; m >>= 1) ss += __shfl_xor(ss, m, 32);
  const float s = 1.0f / fmaxf(sqrtf(ss), 1e-8f);
  float4 o; o.x = v.x * s; o.y = v.y * s; o.z = v.z * s; o.w = v.w * s;
  ((float4*)(zn + (size_t)row * DDIM))[lane] = o;
}

// ---------------------------------------------------------------------------
// Kernel 2: fused sim sweep. Block = 4 waves; wave w owns rows
// r0 = (blockIdx*4 + w)*16, A tile resident in 64 VGPRs. The shared B tile
// (16x128 fp32, contiguous in zn) is double-buffered in LDS and filled with
// GLOBAL_LOAD_ASYNC_TO_LDS_B128 (ASYNCcnt), so no VGPR staging and no spills.
// Per iteration: issue next tile's async copies, run 32 wmma from current
// buffer, fold exp() epilogue, s_wait_asynccnt 0, barrier, flip buffers.
// ---------------------------------------------------------------------------
__global__ __launch_bounds__(TPB) void ntxent_sim_rows(
    const float* __restrict__ zn, float* __restrict__ partials) {
  __shared__ float lb[2][16 * LSTRIDE];

  const int wid  = threadIdx.x >> 5;
  const int lane = threadIdx.x & 31;
  const int lh   = lane & 15;
  const int koff = (lane >> 4) * 2;     // lanes 16-31 carry K+2,K+3
  const int r0   = (blockIdx.x * WAVES + wid) * 16;

  // Resident A tile: rows r0..r0+15, full K=128.
  const float* arow = zn + (size_t)(r0 + lh) * DDIM;
  v2f a[32];
  #pragma unroll
  for (int s = 0; s < 32; ++s)
    a[s] = *(const v2f*)(arow + 4 * s + koff);

  float accS[8], accP[8];
  #pragma unroll
  for (int v = 0; v < 8; ++v) { accS[v] = 0.0f; accP[v] = 0.0f; }

  const int partner0 = (r0 < BHALF) ? (r0 + BHALF) : (r0 - BHALF);
  const int mbase    = (lane >> 4) * 8;   // C layout: row = v + mbase
  const int ncol     = lh;                // C layout: col = lane & 15

  // Per-thread copy geometry: thread t copies float4 elements {t + j*128},
  // j=0..3. Global stride per j = 512 floats; LDS stride per j = 4 rows = 528.
  const int cidx = threadIdx.x;           // float4 index, j=0
  const int crow = cidx >> 5, cq = cidx & 31;
  const int lbase = crow * LSTRIDE + cq * 4;   // float offset in a buffer
  const float* gthr = zn + (size_t)cidx * 4;   // + tile*2048 per tile

  // Stage tile 0 into lb[0].
  #pragma unroll
  for (int j = 0; j < 4; ++j)
    async_copy_b128(gthr + j * 512, &lb[0][lbase + j * 528]);
  wait_async0();
  __syncthreads();

  for (int ct = 0; ct < NTILE; ++ct) {
    const int  cur  = ct & 1;
    const bool more = (ct + 1) < NTILE;

    // Kick off next tile's global->LDS async copies (latency hidden by wmma).
    if (more) {
      const float* g = gthr + (size_t)(ct + 1) * 2048;
      float* l = &lb[cur ^ 1][lbase];
      #pragma unroll
      for (int j = 0; j < 4; ++j)
        async_copy_b128(g + j * 512, l + j * 528);
    }

    // B fragments from LDS (padded stride -> conflict-free ds_load_b64).
    v2f b[32];
    #pragma unroll
    for (int s = 0; s < 32; ++s)
      b[s] = *(const v2f*)&lb[cur][lh * LSTRIDE + 4 * s + koff];

    v8f c = {};
    #pragma unroll
    for (int s = 0; s < 32; ++s) {
      c = __builtin_amdgcn_wmma_f32_16x16x4_f32(
              /*neg_a=*/false, a[s], /*neg_b=*/false, b[s],
              /*c_mod=*/(short)0, c, /*reuse_a=*/false, /*reuse_b=*/false);
    }

    // Epilogue: fold tile into per-row sums; mask self diag; grab positive.
    const int  c0       = ct * 16;
    const bool selfTile = (c0 == r0);
    const bool partTile = (c0 == partner0);
    #pragma unroll
    for (int v = 0; v < 8; ++v) {
      const int  m    = mbase + v;
      const bool diag = (ncol == m);
      const float sim = 2.0f * c[v];                 // cos / TEMPERATURE
      float e = __expf(sim - 2.0f);                  // fixed shift (sim <= 2)
      if (selfTile && diag) e = 0.0f;
      accS[v] += e;
      if (partTile && diag) accP[v] += sim;
    }

    if (more) wait_async0();   // our async writes to lb[cur^1] are in LDS
    __syncthreads();           // make them visible block-wide
  }

  // Reduce across 16-lane column groups: every lane ends with its group's sums
  // (rows mbase..mbase+7), so all lanes do the log/term math redundantly.
  #pragma unroll
  for (int v = 0; v < 8; ++v) {
    #pragma unroll
    for (int m = 8; m >= 1; m >>= 1) {
      accS[v] += __shfl_xor(accS[v], m, 16);
      accP[v] += __shfl_xor(accP[v], m, 16);
    }
  }
  float t = 0.0f;
  #pragma unroll
  for (int v = 0; v < 8; ++v)
    t += __logf(accS[v]) + 2.0f - accP[v];   // lse - pos per row
  t += __shfl_xor(t, 16, 32);                // combine rows 0-7 with 8-15
  if (lane == 0) partials[blockIdx.x * WAVES + wid] = t;
}

// ---------------------------------------------------------------------------
// Kernel 3: reduce the 512 per-wave partials -> loss = sum / NTOT
// ---------------------------------------------------------------------------
__global__ __launch_bounds__(512) void ntxent_finalize(
    const float* __restrict__ partials, float* __restrict__ out) {
  __shared__ float red[16];
  const int tid = threadIdx.x;  // 512 threads = 16 waves
  float v = partials[tid];
  #pragma unroll
  for (int m = 16; m >= 1; m >>= 1) v += __shfl_xor(v, m, 32);
  if ((tid & 31) == 0) red[tid >> 5] = v;
  __syncthreads();
  if (tid < 16) {
    float t = red[tid];
    #pragma unroll
    for (int m = 8; m >= 1; m >>= 1) t += __shfl_xor(t, m, 16);
    if (tid == 0) out[0] = t / (float)NTOT;
  }
}

// ---------------------------------------------------------------------------
extern "C" void kernel_launch(void* const* d_in, const int* in_sizes, int n_in,
                              void* d_out, int out_size, void* d_ws, size_t ws_size,
                              hipStream_t stream) {
  const float* z_i = (const float*)d_in[0];
  const float* z_j = (const float*)d_in[1];
  float* out = (float*)d_out;

  float* zn       = (float*)d_ws;                                   // 4 MB
  float* partials = (float*)((char*)d_ws + (size_t)NTOT * DDIM * sizeof(float));

  ntxent_normalize<<<NTOT / 8, 256, 0, stream>>>(z_i, z_j, zn);
  ntxent_sim_rows<<<NTOT / (16 * WAVES), TPB, 0, stream>>>(zn, partials);
  ntxent_finalize<<<1, 512, 0, stream>>>(partials, out);
}